// Learner_14181982011962
// MI455X (gfx1250) — compile-verified
//
#include <hip/hip_runtime.h>

typedef __attribute__((ext_vector_type(2))) float v2f;
typedef __attribute__((ext_vector_type(8))) float v8f;

// -------------------------------------------------------------------------
// K0: zero the accumulators that are built via atomics (deg, gsum).
// -------------------------------------------------------------------------
__global__ void k_init(float* __restrict__ deg, float* __restrict__ gsum, int n) {
    int i = blockIdx.x * blockDim.x + threadIdx.x;
    if (i < n)  deg[i]  = 0.0f;
    if (i < 16) gsum[i] = 0.0f;
}

// -------------------------------------------------------------------------
// K1: deg[c] += 1 for every edge (self-loop added later as +1).
// -------------------------------------------------------------------------
__global__ void k_degree(const int* __restrict__ col, float* __restrict__ deg, int e) {
    int t = blockIdx.x * blockDim.x + threadIdx.x;
    if (t < e) atomicAdd(&deg[col[t]], 1.0f);
}

// -------------------------------------------------------------------------
// K2: dinv = rsqrt(deg+1); seed scalar accumulator with self-loop x[i]*dinv[i].
// -------------------------------------------------------------------------
__global__ void k_dinv(const float* __restrict__ deg, const float* __restrict__ x,
                       float* __restrict__ dinv, float* __restrict__ sacc, int n) {
    int i = blockIdx.x * blockDim.x + threadIdx.x;
    if (i < n) {
        float di = rsqrtf(deg[i] + 1.0f);   // self-loop => deg >= 1, no zero guard needed
        dinv[i] = di;
        sacc[i] = x[i] * di;
    }
}

// -------------------------------------------------------------------------
// K3: layer-1 aggregation is scalar (in_features == 1): sacc[c] += x[r]*dinv[r].
// -------------------------------------------------------------------------
__global__ void k_scatter1(const int* __restrict__ row, const int* __restrict__ col,
                           const float* __restrict__ x, const float* __restrict__ dinv,
                           float* __restrict__ sacc, int e) {
    int t = blockIdx.x * blockDim.x + threadIdx.x;
    if (t < e) {
        int r = row[t], c = col[t];
        atomicAdd(&sacc[c], x[r] * dinv[r]);
    }
}

// -------------------------------------------------------------------------
// K4: WMMA GEMM. One wave per 16-node tile.
//   h1[m][j] = relu( (dinv[m]*sacc[m]) * W1[j] )
//   p        = diag(dinv) * (h1 @ W2)  computed as  (diag(dinv)*h1) @ W2,
//   i.e. the per-row dinv scale is folded into the A operand, where lane
//   (lane&15) already holds dinv for exactly the A row it supplies.
//   D = A(16x16) @ W2(16x16) via 4x V_WMMA_F32_16X16X4_F32 (full f32).
//   Epilogue stores c[v] directly: no shuffles, no reloads, no guards on the
//   wave-uniform full-tile path (EXEC all-1s throughout, as WMMA requires).
// A layout (16x4 f32): lanes 0-15 = M, VGPR0/1 = K{0,1}; lanes 16-31 carry K{2,3}.
// B layout (4x16 f32): lanes = N, same K striping. C/D: VGPR v holds M=v (+8 hi).
// -------------------------------------------------------------------------
__global__ void k_gemm_wmma(const float* __restrict__ dinv, const float* __restrict__ sacc,
                            const float* __restrict__ W1, const float* __restrict__ W2,
                            float* __restrict__ p, float* __restrict__ acc2,
                            int n, int ntiles) {
    int wave = threadIdx.x >> 5;
    int lane = threadIdx.x & 31;
    int tile = blockIdx.x * (blockDim.x >> 5) + wave;
    if (tile >= ntiles) return;          // wave-uniform: EXEC stays all-1s inside

    int m     = lane & 15;               // row (lanes 0-15) / col index for B
    int hi    = lane >> 4;               // selects K pair {0,1} vs {2,3}
    int base  = tile * 16;
    int nodem = base + m;

    float di   = (nodem < n) ? dinv[nodem] : 0.0f;
    float stot = (nodem < n) ? di * sacc[nodem] : 0.0f;

    v8f c = {0.f, 0.f, 0.f, 0.f, 0.f, 0.f, 0.f, 0.f};
#pragma unroll
    for (int kc = 0; kc < 4; ++kc) {
        int k0 = kc * 4 + hi * 2;
        v2f a, b;
        a.x = di * fmaxf(stot * W1[k0],     0.0f);   // dinv[m]*h1[m][k0]
        a.y = di * fmaxf(stot * W1[k0 + 1], 0.0f);   // dinv[m]*h1[m][k0+1]
        b.x = W2[k0 * 16 + m];                       // W2[k0][n]
        b.y = W2[(k0 + 1) * 16 + m];                 // W2[k0+1][n]
        c = __builtin_amdgcn_wmma_f32_16x16x4_f32(
                /*neg_a=*/false, a, /*neg_b=*/false, b,
                /*c_mod=*/(short)0, c, /*reuse_a=*/false, /*reuse_b=*/false);
    }

    if (base + 16 <= n) {
        // Fast path (always taken for N % 16 == 0): plain coalesced stores.
#pragma unroll
        for (int v = 0; v < 8; ++v) {
            int node = base + v + hi * 8;
            p[node * 16 + m]    = c[v];     // 64B-coalesced per half-wave
            acc2[node * 16 + m] = c[v];     // self-loop seed (p[c] term)
        }
    } else {
        // Partial-tile fallback (unused for this problem size).
#pragma unroll
        for (int v = 0; v < 8; ++v) {
            int node = base + v + hi * 8;
            if (node < n) {
                p[node * 16 + m]    = c[v];
                acc2[node * 16 + m] = c[v];
            }
        }
    }
}

// -------------------------------------------------------------------------
// K5: layer-2 scatter. 16 lanes per edge => p[r*16+j] gathers are 64B blocks.
//     acc2[c][j] += p[r][j]  (dinv[r] already folded into p; dinv[c] applied in K6)
// -------------------------------------------------------------------------
__global__ void k_scatter2(const int* __restrict__ row, const int* __restrict__ col,
                           const float* __restrict__ p, float* __restrict__ acc2,
                           long long total) {
    long long gid = (long long)blockIdx.x * blockDim.x + threadIdx.x;
    if (gid < total) {
        int e = (int)(gid >> 4);
        int j = (int)(gid & 15);
        int r = row[e], c = col[e];
        atomicAdd(&acc2[(long long)c * 16 + j], p[(long long)r * 16 + j]);
    }
}

// -------------------------------------------------------------------------
// K6: readout. h2 = relu(dinv[i]*acc2[i][j]); column sums via LDS + 16 atomics/block.
// blockDim must be 256 so lds[t] with stride 16 shares the feature index j.
// -------------------------------------------------------------------------
__global__ void k_reduce(const float* __restrict__ dinv, const float* __restrict__ acc2,
                         float* __restrict__ gsum, int n) {
    __shared__ float lds[256];
    int gid = blockIdx.x * 256 + threadIdx.x;
    float val = 0.0f;
    if (gid < n * 16) {
        int i = gid >> 4;
        val = fmaxf(dinv[i] * acc2[gid], 0.0f);
    }
    lds[threadIdx.x] = val;
    __syncthreads();
    if (threadIdx.x < 16) {
        float s = 0.0f;
#pragma unroll
        for (int t = 0; t < 256; t += 16) s += lds[threadIdx.x + t];
        atomicAdd(&gsum[threadIdx.x], s);
    }
}

// -------------------------------------------------------------------------
// K7: out[j] = (gsum/N) @ W3 + b   — tiny, one block.
// -------------------------------------------------------------------------
__global__ void k_head(const float* __restrict__ gsum, const float* __restrict__ W3,
                       const float* __restrict__ b, float* __restrict__ out, float inv_n) {
    int j = threadIdx.x;
    if (j < 11) {
        float acc = b[j];
#pragma unroll
        for (int k = 0; k < 16; ++k) acc += (gsum[k] * inv_n) * W3[k * 11 + j];
        out[j] = acc;
    }
}

// -------------------------------------------------------------------------
extern "C" void kernel_launch(void* const* d_in, const int* in_sizes, int n_in,
                              void* d_out, int out_size, void* d_ws, size_t ws_size,
                              hipStream_t stream) {
    const float* x   = (const float*)d_in[0];
    const int*   ei  = (const int*)  d_in[1];   // [2, E] int32
    const float* W1  = (const float*)d_in[2];   // [1,16]
    const float* W2  = (const float*)d_in[3];   // [16,16]
    const float* W3  = (const float*)d_in[4];   // [16,11]
    const float* b   = (const float*)d_in[5];   // [11]
    float* out = (float*)d_out;

    const int n = in_sizes[0];            // 100000
    const int e = in_sizes[1] / 2;        // 3200000
    const int* row = ei;
    const int* col = ei + e;

    // Workspace layout (floats)
    float* ws   = (float*)d_ws;
    float* deg  = ws;                     // [n]
    float* dinv = deg  + n;               // [n]
    float* sacc = dinv + n;               // [n]
    float* gsum = sacc + n;               // [32] (16 used)
    float* p    = gsum + 32;              // [n*16]
    float* acc2 = p    + (size_t)n * 16;  // [n*16]

    const int B = 256;
    const int gN  = (n + B - 1) / B;
    const int gE  = (e + B - 1) / B;
    const int ntiles = (n + 15) / 16;
    const int gT  = (ntiles + 7) / 8;                     // 8 waves/block
    const long long tot2 = (long long)e * 16;
    const int gE16 = (int)((tot2 + B - 1) / B);
    const int gR  = (n * 16 + B - 1) / B;

    k_init    <<<gN,   B, 0, stream>>>(deg, gsum, n);
    k_degree  <<<gE,   B, 0, stream>>>(col, deg, e);
    k_dinv    <<<gN,   B, 0, stream>>>(deg, x, dinv, sacc, n);
    k_scatter1<<<gE,   B, 0, stream>>>(row, col, x, dinv, sacc, e);
    k_gemm_wmma<<<gT,  B, 0, stream>>>(dinv, sacc, W1, W2, p, acc2, n, ntiles);
    k_scatter2<<<gE16, B, 0, stream>>>(row, col, p, acc2, tot2);
    k_reduce  <<<gR,   B, 0, stream>>>(dinv, acc2, gsum, n);
    k_head    <<<1,   32, 0, stream>>>(gsum, W3, b, out, 1.0f / (float)n);
}